// DotAttention_68719477238
// MI455X (gfx1250) — compile-verified
//
#include <hip/hip_runtime.h>
#include <hip/hip_bf16.h>
#include <math.h>

typedef __attribute__((ext_vector_type(2))) float v2f;
typedef __attribute__((ext_vector_type(8))) float v8f;
typedef __attribute__((ext_vector_type(4))) int   v4i;

#define B_ 32
#define S_ 2048
#define D_ 1024
#define BD_ (B_ * D_)
#define T_ (S_ / 16)        // 128 s-tiles per batch
#define PITCH 1028          // 1024 + 4 pad: row stride -> LDS bank stride 4

// gfx1250 async global->LDS staging path (ASYNCcnt-tracked, bypasses VGPRs).
// Guarded so the file still compiles on toolchains without the builtin.
#if defined(__gfx1250__)
#if __has_builtin(__builtin_amdgcn_global_load_async_to_lds_b128) && \
    __has_builtin(__builtin_amdgcn_s_wait_asynccnt)
#define USE_ASYNC_LDS 1
#endif
#endif

// ---------------------------------------------------------------------------
// K1: fused tile pass. One 256-thread block per (16-row s-tile, batch).
//  - stage 16 x 1024 f32 rows of enc into LDS (async b128 loads if available)
//  - energies e_s = <row_s, q> : 8 waves x 128-d slices, WMMA f32 16x16x4
//    with q broadcast across all 16 N columns (GEMV trick; compute is free,
//    the kernel is HBM-bound)
//  - tile-local softmax weights w_s = exp(e_s - m_tile)
//  - partial value acc[d] = sum_s w_s * row_s[d] : WMMA with M=d, K=s
// Writes raw energies (scores region), acc partials + tile max (workspace).
// ---------------------------------------------------------------------------
__global__ __launch_bounds__(256) void dotattn_tile_kernel(
    const float* __restrict__ enc,    // (S, B, D)
    const float* __restrict__ q,      // (B, D)
    float* __restrict__ raw_scores,   // (B, S)  raw energies out
    float* __restrict__ accbuf,       // (B, T_, D) partial weighted sums
    float* __restrict__ mtile)        // (B, T_)  tile maxes
{
    __shared__ float tile[16 * PITCH];   // 65,792 B
    __shared__ float pred[8 * 16];       // per-wave partial energies
    __shared__ float esh[16];            // reduced energies
    __shared__ float wsh[16];            // exp weights

    const int b    = blockIdx.y;
    const int tl   = blockIdx.x;
    const int s0   = tl * 16;
    const int t    = threadIdx.x;
    const int wave = t >> 5;             // 8 waves (wave32)
    const int lane = t & 31;
    const int mrow = lane & 15;          // WMMA M row for this lane
    const int koff = (lane >> 4) * 2;    // lanes 16-31 hold K=2,3

    // ---- stage tile: each iteration, 256 threads cover one full row (4 KB),
    //      perfectly coalesced 128-bit transfers ----
    {
        const float* src = enc + (size_t)s0 * BD_ + (size_t)b * D_;
#if USE_ASYNC_LDS
        for (int i = t; i < 16 * 256; i += 256) {
            const int row = i >> 8;
            const int d   = (i & 255) * 4;
            __builtin_amdgcn_global_load_async_to_lds_b128(
                (__attribute__((address_space(1))) v4i*)(src + (size_t)row * BD_ + d),
                (__attribute__((address_space(3))) v4i*)&tile[row * PITCH + d],
                /*imm offset*/0, /*cpol*/0);
        }
        __builtin_amdgcn_s_wait_asynccnt(0);   // this wave's loads landed
#else
        for (int i = t; i < 16 * 256; i += 256) {
            const int row = i >> 8;
            const int d   = (i & 255) * 4;
            const float4 v = *(const float4*)(src + (size_t)row * BD_ + d);
            *(float4*)(&tile[row * PITCH + d]) = v;
        }
#endif
    }
    __syncthreads();   // all waves' tile rows visible

    // ---- energies: wave w covers d in [w*128, w*128+128) ----
    {
        const float* qrow = q + (size_t)b * D_ + wave * 128 + koff;
        const float* arow = &tile[mrow * PITCH + wave * 128 + koff];
        v8f c = {};
        for (int d = 0; d < 128; d += 4) {
            v2f a  = *(const v2f*)(arow + d);   // ds_load, bank-conflict-free
            v2f bv = *(const v2f*)(qrow + d);   // global (cache-hot)
            c = __builtin_amdgcn_wmma_f32_16x16x4_f32(
                    false, a, false, bv, (short)0, c, false, false);
        }
        // C layout: VGPR r = row M=r (lanes 0-15) / M=r+8 (lanes 16-31);
        // all N columns identical -> lanes 0 and 16 hold 8 results each.
        if (mrow == 0) {
            const int base = wave * 16 + (lane >> 4) * 8;
#pragma unroll
            for (int r = 0; r < 8; ++r) pred[base + r] = c[r];
        }
    }
    __syncthreads();

    if (t < 16) {
        float e = 0.0f;
#pragma unroll
        for (int w = 0; w < 8; ++w) e += pred[w * 16 + t];
        esh[t] = e;
    }
    __syncthreads();
    if (t < 16) {
        float m = esh[0];
#pragma unroll
        for (int i = 1; i < 16; ++i) m = fmaxf(m, esh[i]);
        wsh[t] = expf(esh[t] - m);
        raw_scores[(size_t)b * S_ + s0 + t] = esh[t];
        if (t == 0) mtile[(size_t)b * T_ + tl] = m;
    }
    __syncthreads();

    // ---- weighted accumulation: acc[d] = sum_{s=0..15} w_s * tile[s][d]
    //      WMMA with A[m,k] = tile[k][d0+m], B = w broadcast across N ----
    {
        float* accrow = accbuf + ((size_t)b * T_ + tl) * D_;
        for (int j = 0; j < 8; ++j) {
            const int d0 = wave * 128 + j * 16;
            v8f c = {};
#pragma unroll
            for (int k0 = 0; k0 < 16; k0 += 4) {
                v2f a;
                a.x = tile[(k0 + koff)     * PITCH + d0 + mrow];
                a.y = tile[(k0 + koff + 1) * PITCH + d0 + mrow];
                v2f bv;
                bv.x = wsh[k0 + koff];
                bv.y = wsh[k0 + koff + 1];
                c = __builtin_amdgcn_wmma_f32_16x16x4_f32(
                        false, a, false, bv, (short)0, c, false, false);
            }
            if (mrow == 0) {
                float* dst = accrow + d0 + (lane >> 4) * 8;
#pragma unroll
                for (int r = 0; r < 8; ++r) dst[r] = c[r];
            }
        }
    }
}

// ---------------------------------------------------------------------------
// K2: per-batch softmax in place (max-subtracted, like jax.nn.softmax),
// additionally persisting {rowmax, denom} for the combine pass.
// ---------------------------------------------------------------------------
__global__ __launch_bounds__(256) void dotattn_softmax_kernel(
    float* __restrict__ scores,   // (B, S): raw in / normalized out
    float* __restrict__ Mden)     // (B, 2): {rowmax, denom}
{
    __shared__ float red[256];
    const int b = blockIdx.x;
    const int t = threadIdx.x;
    float* row = scores + (size_t)b * S_;

    float v[S_ / 256];
    float m = -INFINITY;
#pragma unroll
    for (int i = 0; i < S_ / 256; ++i) {
        v[i] = row[t + i * 256];
        m = fmaxf(m, v[i]);
    }
    red[t] = m;
    __syncthreads();
    for (int w = 128; w > 0; w >>= 1) {
        if (t < w) red[t] = fmaxf(red[t], red[t + w]);
        __syncthreads();
    }
    m = red[0];
    __syncthreads();

    float sum = 0.0f;
#pragma unroll
    for (int i = 0; i < S_ / 256; ++i) {
        v[i] = expf(v[i] - m);
        sum += v[i];
    }
    red[t] = sum;
    __syncthreads();
    for (int w = 128; w > 0; w >>= 1) {
        if (t < w) red[t] += red[t + w];
        __syncthreads();
    }
    const float denom = red[0];
    const float inv = 1.0f / denom;
#pragma unroll
    for (int i = 0; i < S_ / 256; ++i) row[t + i * 256] = v[i] * inv;

    if (t == 0) {
        Mden[b * 2 + 0] = m;
        Mden[b * 2 + 1] = denom;
    }
}

// ---------------------------------------------------------------------------
// K3: combine partials: out[b,d] = sum_t exp(m_t - M) * acc[b,t,d] / denom
//   == sum_s softmax(e)_s * enc[s,b,d]. Coalesced: thread = one d.
// ---------------------------------------------------------------------------
__global__ __launch_bounds__(256) void dotattn_combine_kernel(
    const float* __restrict__ accbuf,  // (B, T_, D)
    const float* __restrict__ mtile,   // (B, T_)
    const float* __restrict__ Mden,    // (B, 2)
    float* __restrict__ outv)          // (B, D)
{
    __shared__ float fsh[T_];
    const int b = blockIdx.y;
    const int d = blockIdx.x * 256 + threadIdx.x;
    const float M     = Mden[b * 2 + 0];
    const float denom = Mden[b * 2 + 1];
    if (threadIdx.x < T_)
        fsh[threadIdx.x] = expf(mtile[(size_t)b * T_ + threadIdx.x] - M);
    __syncthreads();

    const float* acc = accbuf + (size_t)b * T_ * D_ + d;
    float s = 0.0f;
    for (int tl = 0; tl < T_; ++tl) s += fsh[tl] * acc[(size_t)tl * D_];
    outv[(size_t)b * D_ + d] = s / denom;
}

// ---------------------------------------------------------------------------
// Launcher. Output layout: [attn_values (B*D) | attn_scores (B*S)].
// Workspace (~16.8 MB): acc partials | tile maxes | per-row {max, denom}.
// Every workspace/output element consumed is produced earlier in the same
// call -> deterministic across replays.
// ---------------------------------------------------------------------------
extern "C" void kernel_launch(void* const* d_in, const int* in_sizes, int n_in,
                              void* d_out, int out_size, void* d_ws, size_t ws_size,
                              hipStream_t stream) {
    const float* q   = (const float*)d_in[0];  // decoder_hidden (1, B, D)
    const float* enc = (const float*)d_in[1];  // encoder_outputs (S, B, D)
    float* out_values = (float*)d_out;                 // (B, D)
    float* out_scores = out_values + (size_t)BD_;      // (B, S)

    float* accbuf = (float*)d_ws;                          // B*T_*D floats
    float* mtile  = accbuf + (size_t)B_ * T_ * D_;         // B*T_ floats
    float* Mden   = mtile + (size_t)B_ * T_;               // B*2 floats

    dotattn_tile_kernel<<<dim3(T_, B_), 256, 0, stream>>>(
        enc, q, out_scores, accbuf, mtile);
    dotattn_softmax_kernel<<<B_, 256, 0, stream>>>(out_scores, Mden);
    dotattn_combine_kernel<<<dim3(D_ / 256, B_), 256, 0, stream>>>(
        accbuf, mtile, Mden, out_values);
}